// GNN_QNetwork_48739288875467
// MI455X (gfx1250) — compile-verified
//
#include <hip/hip_runtime.h>

typedef float v2f __attribute__((ext_vector_type(2)));
typedef float v8f __attribute__((ext_vector_type(8)));

#define N_NODES 100000
#define N_EDGES 1600000
#define IN_CH 32
#define HID 64
#define N_ACT 16

// ---------------------------------------------------------------- utilities
__global__ void k_zero(float* __restrict__ p, int n) {
    int i = blockIdx.x * blockDim.x + threadIdx.x;
    if (i < n) p[i] = 0.0f;
}

// deg[dst] += 1 for each real edge (self-loop added analytically later)
__global__ void k_degree(const int* __restrict__ dst, float* __restrict__ deg, int n_edges) {
    int e = blockIdx.x * blockDim.x + threadIdx.x;
    if (e < n_edges) atomicAdd(&deg[dst[e]], 1.0f);
}

// dinv = rsqrt(deg + 1)   (deg+1 >= 1 always, so no zero-guard needed)
__global__ void k_dinv(float* __restrict__ d, int n) {
    int i = blockIdx.x * blockDim.x + threadIdx.x;
    if (i < n) d[i] = rsqrtf(d[i] + 1.0f);
}

// ---------------------------------------------------------------- WMMA GEMM
// out[N_rows, H] = X[N_rows, K] @ W[K, H] (+bias) (relu)
// fp32 WMMA 16x16x4, one wave per 16x16 output tile.
//
// A (16x4 f32, ISA 7.12.2): lanes 0-15 / 16-31 both hold M=0..15;
//   VGPR0 = K={k0, k0+2} per lane-half, VGPR1 = K={k0+1, k0+3}.
//   => lane half h, row = lane&15, a = {X[row, k0+2h], X[row, k0+2h+1]}
// B (4x16 f32): row-striped across lanes, symmetric with A:
//   => b = {W[k0+2h, col], W[k0+2h+1, col]},  col = lane&15 (+tile)
// C/D (16x16 f32): VGPR i: lanes 0-15 -> M=i, lanes 16-31 -> M=i+8, N=lane&15.
template <int K, int H, bool BIAS, bool RELU>
__global__ void k_gemm_wmma(const float* __restrict__ X, const float* __restrict__ W,
                            const float* __restrict__ bias, float* __restrict__ out,
                            int n_rows) {
    const int NT = H / 16;                       // column tiles
    int wid  = (blockIdx.x * blockDim.x + threadIdx.x) >> 5;
    int mt   = wid / NT;
    int nt   = wid % NT;
    if (mt * 16 >= n_rows) return;               // wave-uniform exit (EXEC stays all-1)

    int lane = threadIdx.x & 31;
    int half = lane >> 4;                        // 0: lanes 0-15, 1: lanes 16-31
    int l16  = lane & 15;

    const float* xrow = X + (size_t)(mt * 16 + l16) * K;   // this lane's A row
    int col = nt * 16 + l16;                               // this lane's B/D column

    v8f acc = {};
#pragma unroll
    for (int k0 = 0; k0 < K; k0 += 4) {
        int ka = k0 + half * 2;
        v2f a = *(const v2f*)(xrow + ka);        // 8B aligned: ka even, row 128B aligned
        v2f b;
        b.x = W[(size_t)ka * H + col];
        b.y = W[(size_t)(ka + 1) * H + col];
        acc = __builtin_amdgcn_wmma_f32_16x16x4_f32(
            /*neg_a=*/false, a, /*neg_b=*/false, b,
            /*c_mod=*/(short)0, acc, /*reuse_a=*/false, /*reuse_b=*/false);
    }

    float bv = BIAS ? bias[col] : 0.0f;
#pragma unroll
    for (int i = 0; i < 8; i++) {
        float v = acc[i] + bv;
        if (RELU) v = fmaxf(v, 0.0f);
        out[(size_t)(mt * 16 + half * 8 + i) * H + col] = v;
    }
}

// ---------------------------------------------------------------- edge scatter
// One wave per edge; lane = channel (2 channels/lane, HID=64).
// agg[dst] += t[src] * dinv[src]*dinv[dst]
__global__ void k_edge_scatter(const int* __restrict__ src, const int* __restrict__ dst,
                               const float* __restrict__ dinv, const float* __restrict__ t,
                               float* __restrict__ agg, int n_edges) {
    int w    = (blockIdx.x * blockDim.x + threadIdx.x) >> 5;
    int lane = threadIdx.x & 31;
    if (w >= n_edges) return;
    int s = src[w];
    int d = dst[w];
    float nrm = dinv[s] * dinv[d];
    const float* ts = t + (size_t)s * HID;
    float*       ad = agg + (size_t)d * HID;
    atomicAdd(ad + lane,      ts[lane]      * nrm);
    atomicAdd(ad + lane + 32, ts[lane + 32] * nrm);
}

// ---------------------------------------------------------------- conv epilogue
// h = relu(agg + t*dinv^2 + bias)   (fuses self-loop contribution + bias + relu)
__global__ void k_conv_finish(const float* __restrict__ agg, const float* __restrict__ t,
                              const float* __restrict__ dinv, const float* __restrict__ bias,
                              float* __restrict__ h, int total) {
    int i = blockIdx.x * blockDim.x + threadIdx.x;
    if (i >= total) return;
    int node = i >> 6;           // / HID
    int c    = i & (HID - 1);
    float di = dinv[node];
    float v  = agg[i] + t[i] * di * di + bias[c];
    h[i] = fmaxf(v, 0.0f);
}

// ---------------------------------------------------------------- launch
extern "C" void kernel_launch(void* const* d_in, const int* in_sizes, int n_in,
                              void* d_out, int out_size, void* d_ws, size_t ws_size,
                              hipStream_t stream) {
    const float* x    = (const float*)d_in[0];
    const int*   ei   = (const int*)d_in[1];       // [2, N_EDGES] row-major
    const float* W_in = (const float*)d_in[2];
    const float* b_in = (const float*)d_in[3];
    const float* W1   = (const float*)d_in[4];
    const float* b1   = (const float*)d_in[5];
    const float* W2   = (const float*)d_in[6];
    const float* b2   = (const float*)d_in[7];
    const float* W_fc = (const float*)d_in[8];
    const float* b_fc = (const float*)d_in[9];
    float* out = (float*)d_out;

    const int* e_src = ei;
    const int* e_dst = ei + N_EDGES;

    char* ws = (char*)d_ws;
    float* dinv = (float*)ws;                                  // N floats
    float* h    = (float*)(ws + (1 << 19));                    // N*HID
    float* t    = h + (size_t)N_NODES * HID;                   // N*HID
    float* agg  = t + (size_t)N_NODES * HID;                   // N*HID

    const int THREADS = 256;
    const int feat_total = N_NODES * HID;

    // 1) degrees -> dinv (re-zeroed every call: graph-replay safe)
    k_zero  <<<(N_NODES + THREADS - 1) / THREADS, THREADS, 0, stream>>>(dinv, N_NODES);
    k_degree<<<(N_EDGES + THREADS - 1) / THREADS, THREADS, 0, stream>>>(e_dst, dinv, N_EDGES);
    k_dinv  <<<(N_NODES + THREADS - 1) / THREADS, THREADS, 0, stream>>>(dinv, N_NODES);

    // 2) input layer: h = relu(x @ W_in + b_in)
    {
        int waves  = (N_NODES / 16) * (HID / 16);
        int blocks = (waves * 32 + THREADS - 1) / THREADS;
        k_gemm_wmma<IN_CH, HID, true, true><<<blocks, THREADS, 0, stream>>>(x, W_in, b_in, h, N_NODES);
    }

    // 3) two GCN conv layers
    const float* Ws[2] = {W1, W2};
    const float* bs[2] = {b1, b2};
    for (int layer = 0; layer < 2; layer++) {
        int waves  = (N_NODES / 16) * (HID / 16);
        int blocks = (waves * 32 + THREADS - 1) / THREADS;
        k_gemm_wmma<HID, HID, false, false><<<blocks, THREADS, 0, stream>>>(h, Ws[layer], nullptr, t, N_NODES);

        k_zero<<<(feat_total + THREADS - 1) / THREADS, THREADS, 0, stream>>>(agg, feat_total);

        int ewaves  = N_EDGES;
        int eblocks = (ewaves * 32 + THREADS - 1) / THREADS;
        k_edge_scatter<<<eblocks, THREADS, 0, stream>>>(e_src, e_dst, dinv, t, agg, N_EDGES);

        k_conv_finish<<<(feat_total + THREADS - 1) / THREADS, THREADS, 0, stream>>>(agg, t, dinv, bs[layer], h, feat_total);
    }

    // 4) output layer: out = h @ W_fc + b_fc
    {
        int waves  = (N_NODES / 16) * (N_ACT / 16);
        int blocks = (waves * 32 + THREADS - 1) / THREADS;
        k_gemm_wmma<HID, N_ACT, true, false><<<blocks, THREADS, 0, stream>>>(h, W_fc, b_fc, out, N_NODES);
    }
}